// MultiPartMemory_20916490731895
// MI455X (gfx1250) — compile-verified
//
#include <hip/hip_runtime.h>
#include <math.h>

typedef __attribute__((ext_vector_type(2))) float v2f;
typedef __attribute__((ext_vector_type(8))) float v8f;

#define P1 4
#define B_  64
#define D_  256
#define N_  4096
#define NEG_K 50
#define INV_TEMP (1.0f/0.07f)
#define BAL_W 0.15f
#define NEG_INF (-1.0e30f)

// ---------------------------------------------------------------------------
// Kernel 1: labels[b] = all_proxy_labels[targets[b]];
//           lfeat[p][b][:] = em[p][labels[b]][:]
// ---------------------------------------------------------------------------
__global__ __launch_bounds__(64) void mpm_gather(const int* __restrict__ targets,
                                                 const int* __restrict__ all_labels,
                                                 const float* __restrict__ em,
                                                 int* __restrict__ labels,
                                                 float* __restrict__ lfeat) {
    const int b = blockIdx.x, p = blockIdx.y, t = threadIdx.x;
    const int lab = all_labels[targets[b]];
    if (p == 0 && t == 0) labels[b] = lab;
    const float4* src = (const float4*)(em + ((size_t)(p * N_ + lab)) * D_);
    float4* dst = (float4*)(lfeat + ((size_t)(p * B_ + b)) * D_);
    dst[t] = src[t];
}

// ---------------------------------------------------------------------------
// Kernel 2: dual GEMM, fp32 WMMA 16x16x4, 2 N-tiles per wave (register block).
// grid (N/32, P1), block 128 (4 waves); wave w owns rows m0=w*16, cols n0..n0+31.
// Per k-step: 4 x b64 loads feed 4 WMMAs (A/L fragments reused across N-tiles,
// B fragments reused across the scores/lsim accumulators).
// ---------------------------------------------------------------------------
__global__ __launch_bounds__(128) void mpm_gemm(const float* __restrict__ feat_global,
                                                const float* __restrict__ feat_part,
                                                const float* __restrict__ em,
                                                const float* __restrict__ lfeat,
                                                float* __restrict__ scaled,
                                                float* __restrict__ sims) {
    const int p    = blockIdx.y;
    const int n0   = blockIdx.x * 32;
    const int wave = threadIdx.x >> 5;
    const int lane = threadIdx.x & 31;
    const int m0   = wave * 16;

    const float* A = (p == 0) ? feat_global : (feat_part + (size_t)(p - 1) * B_ * D_);
    const float* L = lfeat + (size_t)p * B_ * D_;
    const float* E = em + (size_t)p * N_ * D_;

    const int rsel  = lane & 15;
    const int khalf = (lane >> 4) * 2;               // 0 or 2
    const float* arow  = A + (size_t)(m0 + rsel) * D_ + khalf;
    const float* lrow  = L + (size_t)(m0 + rsel) * D_ + khalf;
    const float* brow0 = E + (size_t)(n0 + rsel) * D_ + khalf;
    const float* brow1 = brow0 + (size_t)16 * D_;

    v8f accS0 = {0.f,0.f,0.f,0.f,0.f,0.f,0.f,0.f};
    v8f accS1 = accS0, accL0 = accS0, accL1 = accS0;

#pragma unroll 4
    for (int k = 0; k < D_; k += 4) {
        v2f a  = *(const v2f*)(arow + k);
        v2f l  = *(const v2f*)(lrow + k);
        v2f b0 = *(const v2f*)(brow0 + k);
        v2f b1 = *(const v2f*)(brow1 + k);
        accS0 = __builtin_amdgcn_wmma_f32_16x16x4_f32(false, a, false, b0,
                                                      (short)0, accS0, false, false);
        accS1 = __builtin_amdgcn_wmma_f32_16x16x4_f32(false, a, false, b1,
                                                      (short)0, accS1, false, false);
        accL0 = __builtin_amdgcn_wmma_f32_16x16x4_f32(false, l, false, b0,
                                                      (short)0, accL0, false, false);
        accL1 = __builtin_amdgcn_wmma_f32_16x16x4_f32(false, l, false, b1,
                                                      (short)0, accL1, false, false);
    }

    // C/D layout: VGPR r -> (lanes 0-15: M=r, N=lane) (lanes 16-31: M=r+8, N=lane-16)
    const int col   = n0 + (lane & 15);
    const int mbase = m0 + ((lane >> 4) * 8);
    float* Sout = scaled + (size_t)p * B_ * N_;
    float* Wout = sims   + (size_t)p * B_ * N_;
#pragma unroll
    for (int r = 0; r < 8; ++r) {
        const int m = mbase + r;
        const float s0 = accS0[r], s1 = accS1[r];
        Sout[(size_t)m * N_ + col]      = s0 * INV_TEMP;
        Sout[(size_t)m * N_ + col + 16] = s1 * INV_TEMP;
        Wout[(size_t)m * N_ + col]      = BAL_W * s0 + (1.0f - BAL_W) * accL0[r];
        Wout[(size_t)m * N_ + col + 16] = BAL_W * s1 + (1.0f - BAL_W) * accL1[r];
    }
}

// ------------------- wave32 argmax reduction helper ------------------------
__device__ __forceinline__ void wave_argmax(float& m, int& mi) {
#pragma unroll
    for (int off = 16; off > 0; off >>= 1) {
        const float om = __shfl_down(m, off, 32);
        const int   oi = __shfl_down(mi, off, 32);
        if (om > m) { m = om; mi = oi; }
    }
}

// ---------------------------------------------------------------------------
// Kernel 3: "off" loss per (p,b) row.
// ---------------------------------------------------------------------------
__global__ __launch_bounds__(256) void mpm_off(const float* __restrict__ scaled,
                                               const int* __restrict__ labels,
                                               float* __restrict__ offv) {
    const int b = blockIdx.x, p = blockIdx.y, t = threadIdx.x;
    const int wave = t >> 5, lane = t & 31;
    const float* row = scaled + ((size_t)(p * B_ + b)) * N_;

    __shared__ float v[N_];
    __shared__ float red_v[8];
    __shared__ int   red_i[8];
    __shared__ float posv[4];
    __shared__ float selv[NEG_K];

    for (int i = t; i < N_; i += 256) v[i] = row[i];
    __syncthreads();

    const int base = (labels[b] >> 2) << 2;
    if (t < 4) { posv[t] = v[base + t]; v[base + t] = NEG_INF; }
    __syncthreads();

    for (int it = 0; it < NEG_K; ++it) {
        float m = NEG_INF; int mi = 0;
        for (int i = t; i < N_; i += 256) { float x = v[i]; if (x > m) { m = x; mi = i; } }
        wave_argmax(m, mi);
        if (lane == 0) { red_v[wave] = m; red_i[wave] = mi; }
        __syncthreads();
        if (t == 0) {
            float bm = red_v[0]; int bi = red_i[0];
            for (int w = 1; w < 8; ++w)
                if (red_v[w] > bm) { bm = red_v[w]; bi = red_i[w]; }
            selv[it] = bm;
            v[bi] = NEG_INF;
        }
        __syncthreads();
    }

    if (t == 0) {
        float M = selv[0];
        for (int j = 0; j < 4; ++j) if (posv[j] > M) M = posv[j];
        float ssum = 0.f, ps = 0.f;
        for (int i = 0; i < NEG_K; ++i) ssum += expf(selv[i] - M);
        for (int j = 0; j < 4; ++j) { ssum += expf(posv[j] - M); ps += posv[j]; }
        offv[p * B_ + b] = M + logf(ssum) - 0.25f * ps;
    }
}

// ---------------------------------------------------------------------------
// Kernel 4: "on" loss per (p,b) row.
// Per-camera max: one wave per 512-wide camera chunk (8 waves = 8 cams).
// ---------------------------------------------------------------------------
__global__ __launch_bounds__(256) void mpm_on(const float* __restrict__ scaled,
                                              const float* __restrict__ sims,
                                              float* __restrict__ onv) {
    const int b = blockIdx.x, p = blockIdx.y, t = threadIdx.x;
    const int wave = t >> 5, lane = t & 31;
    const float* srow = scaled + ((size_t)(p * B_ + b)) * N_;
    const float* wrow = sims   + ((size_t)(p * B_ + b)) * N_;

    __shared__ float sv[N_];
    __shared__ float red_v[8];
    __shared__ int   red_i[8];
    __shared__ float camv[8];
    __shared__ int   cami[8];
    __shared__ int   pos2[3];
    __shared__ int   seli[NEG_K];

    for (int i = t; i < N_; i += 256) sv[i] = wrow[i];
    __syncthreads();

    // per-camera max: wave == camera, chunk [wave*512, wave*512+512)
    {
        float m = NEG_INF; int mi = wave * 512;
        for (int i = lane; i < 512; i += 32) {
            const int gi = wave * 512 + i;
            const float x = sv[gi];
            if (x > m) { m = x; mi = gi; }
        }
        wave_argmax(m, mi);
        if (lane == 0) { camv[wave] = m; cami[wave] = mi; }
        __syncthreads();
    }

    // top-3 cameras -> global positive indices; mask them
    if (t == 0) {
        unsigned used = 0;
        for (int j = 0; j < 3; ++j) {
            float bm = NEG_INF; int bc = 0;
            for (int c = 0; c < 8; ++c)
                if (!((used >> c) & 1u) && camv[c] > bm) { bm = camv[c]; bc = c; }
            used |= 1u << bc;
            pos2[j] = cami[bc];
        }
        for (int j = 0; j < 3; ++j) sv[pos2[j]] = NEG_INF;
    }
    __syncthreads();

    // top-50 of remaining sims, keep indices
    for (int it = 0; it < NEG_K; ++it) {
        float m = NEG_INF; int mi = 0;
        for (int i = t; i < N_; i += 256) { float x = sv[i]; if (x > m) { m = x; mi = i; } }
        wave_argmax(m, mi);
        if (lane == 0) { red_v[wave] = m; red_i[wave] = mi; }
        __syncthreads();
        if (t == 0) {
            float bm = red_v[0]; int bi = red_i[0];
            for (int w = 1; w < 8; ++w)
                if (red_v[w] > bm) { bm = red_v[w]; bi = red_i[w]; }
            seli[it] = bi;
            sv[bi] = NEG_INF;
        }
        __syncthreads();
    }

    if (t == 0) {
        float M = NEG_INF, ps = 0.f;
        for (int i = 0; i < NEG_K; ++i) { const float x = srow[seli[i]]; if (x > M) M = x; }
        for (int j = 0; j < 3; ++j) { const float x = srow[pos2[j]]; if (x > M) M = x; ps += x; }
        float ssum = 0.f;
        for (int i = 0; i < NEG_K; ++i) ssum += expf(srow[seli[i]] - M);
        for (int j = 0; j < 3; ++j) ssum += expf(srow[pos2[j]] - M);
        onv[p * B_ + b] = M + logf(ssum) - ps * (1.0f / 3.0f);
    }
}

// ---------------------------------------------------------------------------
// Kernel 5: reduce 2x256 row losses to the 5 outputs.
// ---------------------------------------------------------------------------
__global__ __launch_bounds__(256) void mpm_finalize(const float* __restrict__ offv,
                                                    const float* __restrict__ onv,
                                                    float* __restrict__ out) {
    __shared__ float so[256], sn[256];
    const int t = threadIdx.x;
    so[t] = offv[t];
    sn[t] = onv[t];
    __syncthreads();
    if (t == 0) {
        float goff = 0.f, gon = 0.f, poff = 0.f, pon = 0.f;
        for (int i = 0; i < 64; ++i)  { goff += so[i]; gon += sn[i]; }
        for (int i = 64; i < 256; ++i){ poff += so[i]; pon += sn[i]; }
        goff *= (1.0f / 64.0f);
        gon  *= (1.0f / 64.0f);
        const float poff_l = 0.5f * (poff * (1.0f / 192.0f));
        const float pon_l  = 0.5f * (pon  * (1.0f / 192.0f));
        out[0] = goff + gon + poff_l + pon_l;
        out[1] = goff;
        out[2] = gon;
        out[3] = poff_l;
        out[4] = pon_l;
    }
}

extern "C" void kernel_launch(void* const* d_in, const int* in_sizes, int n_in,
                              void* d_out, int out_size, void* d_ws, size_t ws_size,
                              hipStream_t stream) {
    const float* feat_global = (const float*)d_in[0];  // (64,256)
    const float* feat_part   = (const float*)d_in[1];  // (3,64,256)
    const float* em          = (const float*)d_in[2];  // (4,4096,256)
    const int*   targets     = (const int*)d_in[3];    // (64,)
    const int*   all_labels  = (const int*)d_in[4];    // (16384,)
    float* out = (float*)d_out;                        // 5 scalars

    // workspace (floats): labels(256 pad) | lfeat 65536 | scaled 1M | sims 1M | off 256 | on 256
    float* ws     = (float*)d_ws;
    int*   labels = (int*)ws;
    float* lfeat  = ws + 256;
    float* scaled = lfeat + (size_t)P1 * B_ * D_;
    float* sims   = scaled + (size_t)P1 * B_ * N_;
    float* offv   = sims + (size_t)P1 * B_ * N_;
    float* onv    = offv + 256;

    mpm_gather  <<<dim3(B_, P1),      64, 0, stream>>>(targets, all_labels, em, labels, lfeat);
    mpm_gemm    <<<dim3(N_ / 32, P1), 128, 0, stream>>>(feat_global, feat_part, em, lfeat, scaled, sims);
    mpm_off     <<<dim3(B_, P1),      256, 0, stream>>>(scaled, labels, offv);
    mpm_on      <<<dim3(B_, P1),      256, 0, stream>>>(scaled, sims, onv);
    mpm_finalize<<<1,                 256, 0, stream>>>(offv, onv, out);
}